// SparseMax_VerticalSAMS_78941498900786
// MI455X (gfx1250) — compile-verified
//
#include <hip/hip_runtime.h>
#include <hip/hip_bf16.h>

typedef float v2f __attribute__((ext_vector_type(2)));
typedef float v8f __attribute__((ext_vector_type(8)));
typedef int b128i __attribute__((vector_size(16)));   // matches builtin param type

#define BATCH   8192
#define NFIELD  20
#define SQLEMB  16
#define DATEMB  64
#define HID     1024
#define KEXP    16
#define GIN     (NFIELD * SQLEMB)   // 320
#define EIN     (NFIELD * DATEMB)   // 1280

// -------- CDNA5 async global->LDS staging (probe via __has_builtin) --------
#if defined(__AMDGCN__)
#  if __has_builtin(__builtin_amdgcn_global_load_async_to_lds_b128) && \
      __has_builtin(__builtin_amdgcn_s_wait_asynccnt)
#    define ASYNC_STAGE 1
#  else
#    define ASYNC_STAGE 0
#    warning "gfx1250 async-to-LDS builtins not found; falling back to synchronous LDS staging"
#  endif
#else
#  define ASYNC_STAGE 0
#endif

#if ASYNC_STAGE
#define GL_ASYNC_B128(dst_lds, src_gbl)                                   \
  __builtin_amdgcn_global_load_async_to_lds_b128(                         \
      (__attribute__((address_space(1))) b128i*)(src_gbl),                \
      (__attribute__((address_space(3))) b128i*)(dst_lds), 0, 0)
#else
#define GL_ASYNC_B128(dst_lds, src_gbl)                                   \
  (*(float4*)(dst_lds) = *(const float4*)(src_gbl))
#endif

// ---------------------------------------------------------------- gathers
__global__ void gather_x_kernel(const int* __restrict__ x,
                                const float* __restrict__ tbl,
                                float* __restrict__ x_emb) {
  int t = blockIdx.x * 256 + threadIdx.x;           // B * 320 float4 slots
  if (t >= BATCH * 320) return;
  int b = t / 320, r = t % 320;
  int f = r >> 4, j = (r & 15) << 2;                // 16 float4 per field
  int feat = x[b * NFIELD + f];
  const float4 v = *(const float4*)(tbl + (size_t)feat * DATEMB + j);
  *(float4*)(x_emb + (size_t)b * EIN + f * DATEMB + j) = v;
}

__global__ void gather_sql_kernel(const int* __restrict__ sql,
                                  const float* __restrict__ tbl,
                                  float* __restrict__ sql_emb) {
  int t = blockIdx.x * 256 + threadIdx.x;           // B * 80 float4 slots
  if (t >= BATCH * 80) return;
  int b = t / 80, r = t % 80;
  int f = r >> 2, j = (r & 3) << 2;                 // 4 float4 per field
  int tok = sql[b * NFIELD + f];
  const float4 v = *(const float4*)(tbl + (size_t)tok * SQLEMB + j);
  *(float4*)(sql_emb + (size_t)b * GIN + f * SQLEMB + j) = v;
}

// ------------------------------------------------- generic fp32 WMMA GEMM
// C[M,N] = act(A[M,K] @ W[K,N] + bias[N]); one wave per 16x16 tile.
__global__ void wmma_gemm_bias_kernel(const float* __restrict__ A,
                                      const float* __restrict__ W,
                                      const float* __restrict__ bias,
                                      float* __restrict__ C,
                                      int M, int N, int Kd,
                                      int totalWaves, int doRelu) {
  int wave = (blockIdx.x * blockDim.x + threadIdx.x) >> 5;
  if (wave >= totalWaves) return;                    // wave-uniform
  int lane  = threadIdx.x & 31;
  int ntile = N >> 4;
  int mt = wave / ntile, nt = wave % ntile;

  int m    = lane & 15;                              // A row / C col index
  int half = lane >> 4;
  int koff = half << 1;                              // 0 or 2

  const float* Arow = A + (size_t)(mt * 16 + m) * Kd;
  const float* Wcol = W + nt * 16 + m;

  v8f c = {};
  for (int k0 = 0; k0 < Kd; k0 += 4) {
    v2f a, b;
    a.x = Arow[k0 + koff];
    a.y = Arow[k0 + koff + 1];
    b.x = Wcol[(size_t)(k0 + koff) * N];
    b.y = Wcol[(size_t)(k0 + koff + 1) * N];
    c = __builtin_amdgcn_wmma_f32_16x16x4_f32(false, a, false, b,
                                              (short)0, c, false, false);
  }

  float bn = bias[nt * 16 + m];
#pragma unroll
  for (int v = 0; v < 8; ++v) {
    float val = c[v] + bn;
    if (doRelu) val = fmaxf(val, 0.0f);
    C[(size_t)(mt * 16 + v + half * 8) * N + nt * 16 + m] = val;
  }
}

// ---------------------------------------------------------------- init
__global__ void init_kernel(float* __restrict__ y, float* __restrict__ colsum) {
  int t = blockIdx.x * 256 + threadIdx.x;
  if (t < BATCH) y[t] = 0.0f;
  if (t < KEXP) colsum[t] = 0.0f;
}

// --------------------------------------------------- entmax-1.5 bisection
__global__ void entmax_kernel(const float* __restrict__ g,
                              float* __restrict__ gate,
                              float* __restrict__ colsum) {
  __shared__ float cs[KEXP];
  if (threadIdx.x < KEXP) cs[threadIdx.x] = 0.0f;
  __syncthreads();

  int b = blockIdx.x * 256 + threadIdx.x;
  float xv[KEXP];
  float mx = -1e30f;
#pragma unroll
  for (int i = 0; i < KEXP; ++i) {                   // x = g * (alpha-1)
    xv[i] = g[(size_t)b * KEXP + i] * 0.5f;
    mx = fmaxf(mx, xv[i]);
  }
  float tau_lo = mx - 1.0f;
  float tau_hi = mx - 0.25f;                         // 16^(1-1.5)
  float f_lo = -1.0f;
#pragma unroll
  for (int i = 0; i < KEXP; ++i) {
    float p = fmaxf(xv[i] - tau_lo, 0.0f);
    f_lo += p * p;                                   // p(tau)^(1/(a-1)) = (.)^2
  }
  for (int it = 0; it < 50; ++it) {
    float tau_m = 0.5f * (tau_lo + tau_hi);
    float f_m = -1.0f;
#pragma unroll
    for (int i = 0; i < KEXP; ++i) {
      float p = fmaxf(xv[i] - tau_m, 0.0f);
      f_m += p * p;
    }
    bool same = (f_m * f_lo) >= 0.0f;
    if (same) { tau_lo = tau_m; f_lo = f_m; } else { tau_hi = tau_m; }
  }
  float tau = 0.5f * (tau_lo + tau_hi);
  float pm[KEXP], s = 0.0f;
#pragma unroll
  for (int i = 0; i < KEXP; ++i) {
    float p = fmaxf(xv[i] - tau, 0.0f);
    pm[i] = p * p;
    s += pm[i];
  }
  float inv = 1.0f / s;
#pragma unroll
  for (int i = 0; i < KEXP; ++i) {
    float gs = pm[i] * inv;
    gate[(size_t)b * KEXP + i] = gs;
    atomicAdd(&cs[i], gs);
  }
  __syncthreads();
  if (threadIdx.x < KEXP) atomicAdd(&colsum[threadIdx.x], cs[threadIdx.x]);
}

// ------------------------------------------------------------------ loss
__global__ void loss_kernel(const float* __restrict__ colsum,
                            float* __restrict__ out) {
  if (threadIdx.x != 0 || blockIdx.x != 0) return;
  float mean = 0.0f;
  for (int i = 0; i < KEXP; ++i) mean += colsum[i];
  mean *= (1.0f / KEXP);
  float var = 0.0f;
  for (int i = 0; i < KEXP; ++i) {
    float d = colsum[i] - mean;
    var += d * d;
  }
  var *= (1.0f / (KEXP - 1));                        // unbiased, ddof=1
  out[BATCH] = var / (mean * mean + 1e-10f);
}

// --------------------------------------- fused expert MoE (the hot loop)
// grid = (B/128, K). Workgroup: 4 waves (128 thr), 128 batch rows, 1 expert.
// Each wave owns 32 rows (2 M-subtiles) so B fragments amortize over 2 WMMAs.
// Double-buffered LDS staging via GLOBAL_LOAD_ASYNC_TO_LDS_B128 (ASYNCcnt).
#define MT 128
#define KT 32
#define NSTAGE (EIN / KT)                            // 40
__launch_bounds__(128)
__global__ void expert_fused_kernel(const float* __restrict__ x_emb,
                                    const float* __restrict__ eW1,
                                    const float* __restrict__ eb1,
                                    const float* __restrict__ eW2,
                                    const float* __restrict__ eb2,
                                    const float* __restrict__ gate,
                                    float* __restrict__ y) {
  __shared__ float As[2][MT][36];   // row stride 144B: 16B-aligned, conflict-free
  __shared__ float Bs[2][KT][132];  // row stride 528B: 16B-aligned
  __shared__ float y_s[MT];

  const int b0 = blockIdx.x * MT;
  const int k  = blockIdx.y;
  const int w    = threadIdx.x >> 5;                 // wave id 0..3
  const int lane = threadIdx.x & 31;
  const int m    = lane & 15;
  const int half = lane >> 4;
  const int koff = half << 1;

  const float* eW1k = eW1 + (size_t)k * EIN * HID;
  const float* eb1k = eb1 + (size_t)k * HID;
  const float* eW2k = eW2 + (size_t)k * HID;

  // stage K-slab s of eW1[:, nb:nb+128] and x_emb rows into buffer `buf`
  auto stage = [&](int s, int buf, int nb) {
    const int k0 = s * KT;
#pragma unroll
    for (int i = 0; i < 8; ++i) {
      int e = threadIdx.x + i * 128;
      {
        int row = e >> 3, c4 = (e & 7) << 2;         // A: 128x32 = 1024 f4
        const float* src = &x_emb[(size_t)(b0 + row) * EIN + k0 + c4];
        float* dst = &As[buf][row][c4];
        GL_ASYNC_B128(dst, src);
      }
      {
        int row = e >> 5, c4 = (e & 31) << 2;        // B: 32x128 = 1024 f4
        const float* src = &eW1k[(size_t)(k0 + row) * HID + nb + c4];
        float* dst = &Bs[buf][row][c4];
        GL_ASYNC_B128(dst, src);
      }
    }
  };

  float part0[8], part1[8];
#pragma unroll
  for (int v = 0; v < 8; ++v) { part0[v] = 0.0f; part1[v] = 0.0f; }

  for (int nb = 0; nb < HID; nb += 128) {            // 8 column blocks
    v8f acc0[8], acc1[8];
#pragma unroll
    for (int t = 0; t < 8; ++t) { acc0[t] = (v8f){}; acc1[t] = (v8f){}; }

    stage(0, 0, nb);
    for (int s = 0; s < NSTAGE; ++s) {
      const int buf = s & 1;
      if (s + 1 < NSTAGE) {
        stage(s + 1, buf ^ 1, nb);
#if ASYNC_STAGE
        __builtin_amdgcn_s_wait_asynccnt(16);        // stage s landed, s+1 in flight
#endif
      } else {
#if ASYNC_STAGE
        __builtin_amdgcn_s_wait_asynccnt(0);
#endif
      }
      __syncthreads();

#pragma unroll
      for (int kk = 0; kk < KT; kk += 4) {
        v2f a0, a1;
        a0.x = As[buf][32 * w + m][kk + koff];
        a0.y = As[buf][32 * w + m][kk + koff + 1];
        a1.x = As[buf][32 * w + 16 + m][kk + koff];
        a1.y = As[buf][32 * w + 16 + m][kk + koff + 1];
#pragma unroll
        for (int t = 0; t < 8; ++t) {
          v2f bfr;
          bfr.x = Bs[buf][kk + koff][16 * t + m];
          bfr.y = Bs[buf][kk + koff + 1][16 * t + m];
          acc0[t] = __builtin_amdgcn_wmma_f32_16x16x4_f32(
              false, a0, false, bfr, (short)0, acc0[t], false, false);
          acc1[t] = __builtin_amdgcn_wmma_f32_16x16x4_f32(
              false, a1, false, bfr, (short)0, acc1[t], false, false);
        }
      }
      __syncthreads();
    }

    // epilogue for this column block: bias + relu + dot with eW2, fused
#pragma unroll
    for (int t = 0; t < 8; ++t) {
      int col = nb + 16 * t + m;
      float b1 = eb1k[col];
      float w2 = eW2k[col];
#pragma unroll
      for (int v = 0; v < 8; ++v) {
        part0[v] += fmaxf(acc0[t][v] + b1, 0.0f) * w2;
        part1[v] += fmaxf(acc1[t][v] + b1, 0.0f) * w2;
      }
    }
  }

  // reduce over the 16 lanes of each half (the N dimension of C tiles)
#pragma unroll
  for (int v = 0; v < 8; ++v) {
    float r0 = part0[v];
    r0 += __shfl_xor(r0, 1);
    r0 += __shfl_xor(r0, 2);
    r0 += __shfl_xor(r0, 4);
    r0 += __shfl_xor(r0, 8);
    float r1 = part1[v];
    r1 += __shfl_xor(r1, 1);
    r1 += __shfl_xor(r1, 2);
    r1 += __shfl_xor(r1, 4);
    r1 += __shfl_xor(r1, 8);
    if (lane == 0)  { y_s[32 * w + v]      = r0; y_s[32 * w + 16 + v] = r1; }
    if (lane == 16) { y_s[32 * w + 8 + v]  = r0; y_s[32 * w + 24 + v] = r1; }
  }
  __syncthreads();

  {
    int b = b0 + threadIdx.x;                        // 128 threads, 128 rows
    float yk = y_s[threadIdx.x] + eb2[k];
    float gs = gate[(size_t)b * KEXP + k];
    atomicAdd(&y[b], gs * yk);
  }
}

// ---------------------------------------------------------------- launch
extern "C" void kernel_launch(void* const* d_in, const int* in_sizes, int n_in,
                              void* d_out, int out_size, void* d_ws, size_t ws_size,
                              hipStream_t stream) {
  const int*   x       = (const int*)d_in[0];
  const int*   sql     = (const int*)d_in[1];
  const float* sql_tbl = (const float*)d_in[2];
  const float* in_tbl  = (const float*)d_in[3];
  const float* gW1     = (const float*)d_in[4];
  const float* gb1     = (const float*)d_in[5];
  const float* gW2     = (const float*)d_in[6];
  const float* gb2     = (const float*)d_in[7];
  const float* eW1     = (const float*)d_in[8];
  const float* eb1     = (const float*)d_in[9];
  const float* eW2     = (const float*)d_in[10];
  const float* eb2     = (const float*)d_in[11];
  float* out = (float*)d_out;

  float* ws = (float*)d_ws;
  float* xe     = ws;                                   // 8192*1280
  float* se     = xe + (size_t)BATCH * EIN;             // 8192*320
  float* hg     = se + (size_t)BATCH * GIN;             // 8192*1024
  float* g      = hg + (size_t)BATCH * HID;             // 8192*16
  float* gate   = g  + (size_t)BATCH * KEXP;            // 8192*16
  float* colsum = gate + (size_t)BATCH * KEXP;          // 16

  gather_x_kernel  <<<(BATCH * 320) / 256, 256, 0, stream>>>(x, in_tbl, xe);
  gather_sql_kernel<<<(BATCH * 80)  / 256, 256, 0, stream>>>(sql, sql_tbl, se);

  int waves1 = (BATCH / 16) * (HID / 16);               // 32768
  wmma_gemm_bias_kernel<<<waves1 / 8, 256, 0, stream>>>(
      se, gW1, gb1, hg, BATCH, HID, GIN, waves1, 1);

  int waves2 = (BATCH / 16) * (KEXP / 16);              // 512
  wmma_gemm_bias_kernel<<<waves2 / 8, 256, 0, stream>>>(
      hg, gW2, gb2, g, BATCH, KEXP, HID, waves2, 0);

  init_kernel  <<<BATCH / 256, 256, 0, stream>>>(out, colsum);
  entmax_kernel<<<BATCH / 256, 256, 0, stream>>>(g, gate, colsum);

  dim3 egrid(BATCH / MT, KEXP);                         // 64 x 16
  expert_fused_kernel<<<egrid, 128, 0, stream>>>(xe, eW1, eb1, eW2, eb2, gate, out);

  loss_kernel<<<1, 32, 0, stream>>>(colsum, out);
}